// LatentDecoder_50560355008867
// MI455X (gfx1250) — compile-verified
//
#include <hip/hip_runtime.h>
#include <cstdint>

#define NNODE 10000
#define NEDGE 160000
#define CHK   20000   // edge chunk (multiple of 16, divides E)

typedef float v2f __attribute__((ext_vector_type(2)));
typedef float v8f __attribute__((ext_vector_type(8)));

__device__ __forceinline__ float siluf(float v) { return v * (1.f / (1.f + __expf(-v))); }

// ---------------------------------------------------------------------------
// Generic fp32 WMMA GEMM:  C[M x Ncols] = act(alpha*A@B (+C if accum) (+bias))
// A: lda elements/row, B: K x Ncols row-major, C: ldc elements/row.
// Tile: 16 (M) x 64 (N) per 128-thread (4-wave) workgroup, K chunked by 64.
// Uses V_WMMA_F32_16X16X4_F32 (wave32, fp32 in/out).
// ---------------------------------------------------------------------------
__launch_bounds__(128)
__global__ void k_gemm(const float* __restrict__ A, int lda,
                       const float* __restrict__ B, int Ncols, int K,
                       float* __restrict__ C, int ldc, int M,
                       const float* __restrict__ bias, float alpha,
                       int accum, int act)
{
  __shared__ float As[16][65];
  __shared__ float Bs[64][68];
  const int m0 = blockIdx.x * 16;
  const int n0 = blockIdx.y * 64;
  const int tid  = threadIdx.x;
  const int wave = tid >> 5;
  const int lane = tid & 31;
  const int hlf  = lane >> 4;      // 0: lanes 0-15, 1: lanes 16-31
  const int lm   = lane & 15;
  v8f acc = {0.f,0.f,0.f,0.f,0.f,0.f,0.f,0.f};

  for (int k0 = 0; k0 < K; k0 += 64) {
    for (int i = tid; i < 16 * 64; i += 128) {
      int r = i >> 6, c = i & 63;
      int kg = k0 + c, mg = m0 + r;
      As[r][c] = (kg < K && mg < M) ? A[(size_t)mg * lda + kg] : 0.f;
    }
    for (int i = tid; i < 64 * 64; i += 128) {
      int r = i >> 6, c = i & 63;
      int kg = k0 + r, ng = n0 + c;
      Bs[r][c] = (kg < K && ng < Ncols) ? B[(size_t)kg * Ncols + ng] : 0.f;
    }
    __syncthreads();
    const int nn = wave * 16 + lm;
#pragma unroll
    for (int kk = 0; kk < 64; kk += 4) {
      v2f a, b;
      a.x = As[lm][kk + 2 * hlf];
      a.y = As[lm][kk + 2 * hlf + 1];
      b.x = Bs[kk + 2 * hlf][nn];
      b.y = Bs[kk + 2 * hlf + 1][nn];
      acc = __builtin_amdgcn_wmma_f32_16x16x4_f32(false, a, false, b,
                                                  (short)0, acc, false, false);
    }
    __syncthreads();
  }

#pragma unroll
  for (int i = 0; i < 8; ++i) {
    int m = m0 + i + 8 * hlf;
    int n = n0 + wave * 16 + lm;
    if (m < M && n < Ncols) {
      float v = alpha * acc[i];
      if (accum) v += C[(size_t)m * ldc + n];
      if (bias)  v += bias[n];
      if (act == 1)      v = siluf(v);
      else if (act == 2) v = fmaxf(v, 0.f);
      C[(size_t)m * ldc + n] = v;
    }
  }
}

// ---------------------------------------------------------------------------
// Elementwise / graph kernels
// ---------------------------------------------------------------------------
__global__ void k_zero(float* p, long n) {
  long i = (long)blockIdx.x * blockDim.x + threadIdx.x;
  if (i < n) p[i] = 0.f;
}

__global__ void k_edge_rot(const float* __restrict__ x, const int* __restrict__ ei,
                           float* __restrict__ D) {
  long e = (long)blockIdx.x * blockDim.x + threadIdx.x;
  if (e >= NEDGE) return;
  int s = ei[e], d = ei[NEDGE + e];
  float vx = x[d*3+0]-x[s*3+0], vy = x[d*3+1]-x[s*3+1], vz = x[d*3+2]-x[s*3+2];
  float inv = 1.f / (sqrtf(vx*vx+vy*vy+vz*vz) + 1e-8f);
  vx*=inv; vy*=inv; vz*=inv;
  float rx, ry, rz;
  if (fabsf(vz) < 0.99f) { rx=0.f; ry=0.f; rz=1.f; } else { rx=1.f; ry=0.f; rz=0.f; }
  float ux = vy*rz - vz*ry, uy = vz*rx - vx*rz, uz = vx*ry - vy*rx;
  float ui = 1.f / (sqrtf(ux*ux+uy*uy+uz*uz) + 1e-8f);
  ux*=ui; uy*=ui; uz*=ui;
  float wx = vy*uz - vz*uy, wy = vz*ux - vx*uz, wz = vx*uy - vy*ux;
  float R[3][3] = {{ux,uy,uz},{vx,vy,vz},{wx,wy,wz}};
  const int p[3] = {1,2,0};
  float* o = D + (size_t)e * 9;
  for (int i = 0; i < 3; ++i)
    for (int j = 0; j < 3; ++j)
      o[i*3+j] = R[p[i]][p[j]];
}

__global__ void k_build_bbf(const float* __restrict__ bb_s, const float* __restrict__ bb_v,
                            float* __restrict__ bbf) {
  long idx = (long)blockIdx.x * blockDim.x + threadIdx.x;
  if (idx >= (long)NNODE * 24) return;
  int n = idx / 24, r = idx % 24, c = r / 6, k = r % 6;
  bbf[idx] = (c == 0) ? bb_s[n*6 + k] : bb_v[(size_t)n*18 + k*3 + (c-1)];
}

// cat for _conv (src only): per-edge row [c0|c2|c1|c3], each cin wide.
__global__ void k_pack_cat_conv(const float* __restrict__ xf, const float* __restrict__ gatec,
                                const float* __restrict__ D, const int* __restrict__ ei,
                                int e0, int CHc, float* __restrict__ catc, int cin) {
  long idx = (long)blockIdx.x * blockDim.x + threadIdx.x;
  if (idx >= (long)CHc * cin) return;
  int el = idx / cin, k = idx % cin;
  int e = e0 + el;
  int node = ei[e];
  const float* xb = xf + (size_t)node * 4 * cin;
  float s  = xb[k];
  float v0 = xb[cin + k], v1 = xb[2*cin + k], v2 = xb[3*cin + k];
  const float* d = D + (size_t)e * 9;
  float r0 = d[0]*v0 + d[1]*v1 + d[2]*v2;
  float r1 = d[3]*v0 + d[4]*v1 + d[5]*v2;
  float r2 = d[6]*v0 + d[7]*v1 + d[8]*v2;
  float g = gatec[(size_t)el * cin + k];
  float* row = catc + (size_t)el * 4 * cin;
  row[k]           = s  * g;   // c0
  row[cin + k]     = r1 * g;   // c2
  row[2*cin + k]   = r0 * g;   // c1
  row[3*cin + k]   = r2 * g;   // c3
}

// cat for _block (src||dst): per-edge row [c0|c2|c1|c3], each 128 wide.
__global__ void k_pack_cat_block(const float* __restrict__ xn, const float* __restrict__ gatec,
                                 const float* __restrict__ D, const int* __restrict__ ei,
                                 int e0, int CHc, float* __restrict__ catc) {
  long idx = (long)blockIdx.x * blockDim.x + threadIdx.x;
  if (idx >= (long)CHc * 128) return;
  int el = idx >> 7, k = idx & 127;
  int e = e0 + el;
  int node = (k < 64) ? ei[e] : ei[NEDGE + e];
  int kk = k & 63;
  const float* xb = xn + (size_t)node * 256;
  float s  = xb[kk];
  float v0 = xb[64 + kk], v1 = xb[128 + kk], v2 = xb[192 + kk];
  const float* d = D + (size_t)e * 9;
  float r0 = d[0]*v0 + d[1]*v1 + d[2]*v2;
  float r1 = d[3]*v0 + d[4]*v1 + d[5]*v2;
  float r2 = d[6]*v0 + d[7]*v1 + d[8]*v2;
  float g = gatec[(size_t)el * 128 + k];
  float* row = catc + (size_t)el * 512;
  row[k]        = s  * g;
  row[128 + k]  = r1 * g;
  row[256 + k]  = r0 * g;
  row[384 + k]  = r2 * g;
}

__global__ void k_logit(const float* __restrict__ af, const float* __restrict__ avec,
                        float* __restrict__ logit, int CHc) {
  long idx = (long)blockIdx.x * blockDim.x + threadIdx.x;
  if (idx >= (long)CHc * 8) return;
  int el = idx >> 3, h = idx & 7;
  const float* a = af + (size_t)el * 128 + h * 16;
  const float* w = avec + h * 16;
  float s = 0.f;
#pragma unroll
  for (int i = 0; i < 16; ++i) s += a[i] * w[i];
  logit[idx] = s;
}

__global__ void k_init_stats(float* mx, float* den) {
  long i = (long)blockIdx.x * blockDim.x + threadIdx.x;
  if (i >= (long)NNODE * 8) return;
  mx[i] = -1e30f; den[i] = 0.f;
}

__device__ __forceinline__ void atomicMaxF(float* addr, float val) {
  unsigned int* a = (unsigned int*)addr;
  unsigned int old = *a;
  while (true) {
    float f = __uint_as_float(old);
    if (f >= val) break;
    unsigned int assumed = old;
    old = atomicCAS(a, assumed, __float_as_uint(val));
    if (old == assumed) break;
  }
}

__global__ void k_amax(const float* __restrict__ logit, const int* __restrict__ ei,
                       float* __restrict__ mx) {
  long idx = (long)blockIdx.x * blockDim.x + threadIdx.x;
  if (idx >= (long)NEDGE * 8) return;
  int e = idx >> 3, h = idx & 7;
  atomicMaxF(&mx[(size_t)ei[NEDGE + e] * 8 + h], logit[idx]);
}

__global__ void k_expsum(const float* __restrict__ logit, const int* __restrict__ ei,
                         const float* __restrict__ mx, float* __restrict__ den,
                         float* __restrict__ wx) {
  long idx = (long)blockIdx.x * blockDim.x + threadIdx.x;
  if (idx >= (long)NEDGE * 8) return;
  int e = idx >> 3, h = idx & 7;
  int d = ei[NEDGE + e];
  float w = __expf(logit[idx] - mx[(size_t)d * 8 + h]);
  wx[idx] = w;
  atomicAdd(&den[(size_t)d * 8 + h], w);
}

__global__ void k_attndiv(float* __restrict__ wx, const int* __restrict__ ei,
                          const float* __restrict__ den) {
  long idx = (long)blockIdx.x * blockDim.x + threadIdx.x;
  if (idx >= (long)NEDGE * 8) return;
  int e = idx >> 3, h = idx & 7;
  wx[idx] /= den[(size_t)ei[NEDGE + e] * 8 + h];
}

// attention-weighted rot_inv scatter for _block (cout=HV=64)
__global__ void k_scatter_block(const float* __restrict__ eA, const float* __restrict__ eB,
                                const float* __restrict__ eC, const float* __restrict__ attn,
                                const float* __restrict__ D, const int* __restrict__ ei,
                                float* __restrict__ agg) {
  long idx = (long)blockIdx.x * blockDim.x + threadIdx.x;
  if (idx >= (long)NEDGE * 256) return;
  int e = idx >> 8, r = idx & 255, c = r >> 6, n = r & 63;
  int dst = ei[NEDGE + e];
  float a = attn[(size_t)e * 8 + (n >> 3)];
  float val;
  if (c == 0) {
    val = a * eA[(size_t)e * 128 + n];              // ch0 = y0[:, :64]
  } else {
    int i = c - 1;
    float v1 = eC[(size_t)e * 64 + n];              // ch1 = yi
    float v2 = eA[(size_t)e * 128 + 64 + n];        // ch2 = y0[:, 64:]
    float v3 = eB[(size_t)e * 64 + n];              // ch3 = yr
    const float* d = D + (size_t)e * 9;
    val = a * (d[0*3+i]*v1 + d[1*3+i]*v2 + d[2*3+i]*v3);
  }
  atomicAdd(&agg[(size_t)dst * 256 + c * 64 + n], val);
}

// rot_inv scatter for _conv with 1/sqrt(deg)=0.25 (chunked)
__global__ void k_scatter_conv(const float* __restrict__ y0c, const float* __restrict__ yrc,
                               const float* __restrict__ yic, const float* __restrict__ D,
                               const int* __restrict__ ei, int e0, int CHc,
                               float* __restrict__ agg, int cout) {
  long idx = (long)blockIdx.x * blockDim.x + threadIdx.x;
  if (idx >= (long)CHc * 4 * cout) return;
  int el = idx / (4 * cout), r = idx % (4 * cout), c = r / cout, n = r % cout;
  int e = e0 + el;
  int dst = ei[NEDGE + e];
  float val;
  if (c == 0) {
    val = y0c[(size_t)el * 2 * cout + n];
  } else {
    int i = c - 1;
    float v1 = yic[(size_t)el * cout + n];
    float v2 = y0c[(size_t)el * 2 * cout + cout + n];
    float v3 = yrc[(size_t)el * cout + n];
    const float* d = D + (size_t)e * 9;
    val = d[0*3+i]*v1 + d[1*3+i]*v2 + d[2*3+i]*v3;
  }
  atomicAdd(&agg[(size_t)dst * 4 * cout + c * cout + n], 0.25f * val);
}

__global__ void k_eqnorm(const float* __restrict__ in, float* __restrict__ out,
                         const float* __restrict__ g0, const float* __restrict__ b0,
                         const float* __restrict__ g1, int cin) {
  int n = blockIdx.x * blockDim.x + threadIdx.x;
  if (n >= NNODE) return;
  const float* x = in + (size_t)n * 4 * cin;
  float* o = out + (size_t)n * 4 * cin;
  float mu = 0.f;
  for (int k = 0; k < cin; ++k) mu += x[k];
  mu /= (float)cin;
  float var = 0.f;
  for (int k = 0; k < cin; ++k) { float d = x[k] - mu; var += d * d; }
  float sd = rsqrtf(var / (float)cin + 1e-5f);
  float vs = 0.f;
  for (int k = cin; k < 4 * cin; ++k) vs += x[k] * x[k];
  float vn = rsqrtf(vs / (float)(3 * cin) + 1e-5f);
  for (int k = 0; k < cin; ++k) o[k] = (x[k] - mu) * sd * g0[k] + b0[k];
  for (int c = 1; c < 4; ++c)
    for (int k = 0; k < cin; ++k)
      o[c*cin + k] = x[c*cin + k] * vn * g1[k];
}

__global__ void k_add(float* a, const float* b, long n) {
  long i = (long)blockIdx.x * blockDim.x + threadIdx.x;
  if (i < n) a[i] += b[i];
}

__global__ void k_vvmul(float* __restrict__ t4, const float* __restrict__ ng) {
  long idx = (long)blockIdx.x * blockDim.x + threadIdx.x;
  if (idx >= (long)NNODE * 128) return;
  t4[idx] *= ng[(idx >> 7) * 32 + (idx & 31)];
}

__global__ void k_assemble(const float* __restrict__ nA, const float* __restrict__ nhs,
                           const float* __restrict__ hv4, float* __restrict__ outp, int cout) {
  long idx = (long)blockIdx.x * blockDim.x + threadIdx.x;
  if (idx >= (long)NNODE * 4 * cout) return;
  int n = idx / (4 * cout), r = idx % (4 * cout), c = r / cout, k = r % cout;
  outp[idx] = nA[idx] + ((c == 0) ? nhs[(size_t)n * cout + k] : hv4[idx]);
}

__global__ void k_combine_embed(const float* __restrict__ agg, const float* __restrict__ latent,
                                float* __restrict__ res) {
  long idx = (long)blockIdx.x * blockDim.x + threadIdx.x;
  if (idx >= (long)NNODE * 256) return;
  int n = idx >> 8, r = idx & 255, c = r >> 6, k = r & 63;
  res[idx] = (k < 32) ? agg[(size_t)n * 128 + c * 32 + k]
                      : latent[(size_t)n * 128 + c * 32 + (k - 32)];
}

__global__ void k_pack_z(const float* __restrict__ e, const float* __restrict__ res,
                         const int* __restrict__ ei, int e0, int CHc, float* __restrict__ zc) {
  long idx = (long)blockIdx.x * blockDim.x + threadIdx.x;
  if (idx >= (long)CHc * 256) return;
  int el = idx >> 8, k = idx & 255;
  int eg = e0 + el;
  float v;
  if (k < 128)      v = e[(size_t)eg * 128 + k];
  else if (k < 192) v = res[(size_t)ei[eg] * 256 + (k - 128)];
  else              v = res[(size_t)ei[NEDGE + eg] * 256 + (k - 192)];
  zc[idx] = v;
}

__global__ void k_seq_ln(const float* __restrict__ atom, const float* __restrict__ lg,
                         const float* __restrict__ lb, float* __restrict__ hh) {
  int n = blockIdx.x * blockDim.x + threadIdx.x;
  if (n >= NNODE) return;
  const float* s = atom + (size_t)n * 364;  // channel 0 of (4,91)
  float mu = 0.f;
  for (int k = 0; k < 91; ++k) mu += s[k];
  mu /= 91.f;
  float var = 0.f;
  for (int k = 0; k < 91; ++k) { float d = s[k] - mu; var += d * d; }
  float sd = rsqrtf(var / 91.f + 1e-5f);
  for (int k = 0; k < 91; ++k) hh[(size_t)n * 91 + k] = (s[k] - mu) * sd * lg[k] + lb[k];
}

__global__ void k_logsoftmax(const float* __restrict__ sl, float* __restrict__ outp) {
  int n = blockIdx.x * blockDim.x + threadIdx.x;
  if (n >= NNODE) return;
  const float* v = sl + (size_t)n * 20;
  float mx = v[0];
  for (int k = 1; k < 20; ++k) mx = fmaxf(mx, v[k]);
  float s = 0.f;
  for (int k = 0; k < 20; ++k) s += __expf(v[k] - mx);
  float ls = logf(s);
  for (int k = 0; k < 20; ++k) outp[(size_t)n * 20 + k] = v[k] - mx - ls;
}

__global__ void k_decoded(const float* __restrict__ atom, float* __restrict__ outp) {
  long idx = (long)blockIdx.x * blockDim.x + threadIdx.x;
  if (idx >= (long)NNODE * 273) return;
  int n = idx / 273, r = idx % 273, a = r / 3, i = r % 3;
  outp[idx] = atom[(size_t)n * 364 + (1 + i) * 91 + a];
}

// ---------------------------------------------------------------------------
// Host side
// ---------------------------------------------------------------------------
static inline dim3 g1(long n) { return dim3((unsigned)((n + 255) / 256)); }

static inline void gemm(hipStream_t st, const float* A, int lda, const float* B,
                        int Ncols, int K, float* C, int ldc, int M,
                        const float* bias, float alpha, int accum, int act) {
  dim3 grid((unsigned)((M + 15) / 16), (unsigned)((Ncols + 63) / 64));
  k_gemm<<<grid, 128, 0, st>>>(A, lda, B, Ncols, K, C, ldc, M, bias, alpha, accum, act);
}

struct ConvP { const float *W0, *Wr, *Wi, *We1, *be1, *We2, *be2; };

extern "C" void kernel_launch(void* const* d_in, const int* in_sizes, int n_in,
                              void* d_out, int out_size, void* d_ws, size_t ws_size,
                              hipStream_t stream)
{
  (void)in_sizes; (void)n_in; (void)out_size; (void)ws_size;
  const int N = NNODE, E = NEDGE;
  const float* x      = (const float*)d_in[0];
  const float* bb_s   = (const float*)d_in[1];
  const float* bb_v   = (const float*)d_in[2];
  const float* latent = (const float*)d_in[3];
  const float* edge_s = (const float*)d_in[4];
  const int*   ei     = (const int*)  d_in[5];
  auto P = [&](int i) { return (const float*)d_in[i]; };
  auto conv_at = [&](int b) {
    ConvP c; c.W0 = P(b); c.Wr = P(b+1); c.Wi = P(b+2);
    c.We1 = P(b+3); c.be1 = P(b+4); c.We2 = P(b+5); c.be2 = P(b+6); return c;
  };

  // ---- workspace carve (floats) ----
  float* W = (float*)d_ws;
  size_t o = 0;
  auto alloc = [&](size_t n) { float* p = W + o; o += (n + 63) & ~(size_t)63; return p; };
  float* Dm     = alloc((size_t)E * 9);
  float* eCur   = alloc((size_t)E * 128);
  float* res    = alloc((size_t)N * 256);
  float* xn     = alloc((size_t)N * 256);
  float* nA     = alloc((size_t)N * 256);
  float* agg    = alloc((size_t)N * 256);
  float* nB     = alloc((size_t)N * 128);
  float* nh1    = alloc((size_t)N * 32);
  float* ng     = alloc((size_t)N * 32);
  float* nhs    = alloc((size_t)N * 64);
  float* atom   = alloc((size_t)N * 364);
  float* eA     = alloc((size_t)E * 128);
  float* eB     = alloc((size_t)E * 64);
  float* eC     = alloc((size_t)E * 64);
  float* logitb = alloc((size_t)E * 8);
  float* attnb  = alloc((size_t)E * 8);
  float* mxb    = alloc((size_t)N * 8);
  float* denb   = alloc((size_t)N * 8);
  float* bbf    = alloc((size_t)N * 24);
  float* catc   = alloc((size_t)CHK * 512);   // reused as z-pack (CHK*256)
  float* gatec  = alloc((size_t)CHK * 128);
  float* h1c    = alloc((size_t)CHK * 64);
  float* y0c    = alloc((size_t)CHK * 192);
  float* yrc    = alloc((size_t)CHK * 96);
  float* yic    = alloc((size_t)CHK * 96);
  float* hh     = alloc((size_t)N * 91);
  float* hr     = alloc((size_t)N * 91);
  float* sl     = alloc((size_t)N * 20);

  hipMemcpyAsync(eCur, edge_s, (size_t)E * 128 * 4, hipMemcpyDeviceToDevice, stream);
  k_edge_rot<<<g1(E), 256, 0, stream>>>(x, ei, Dm);
  k_build_bbf<<<g1((long)N * 24), 256, 0, stream>>>(bb_s, bb_v, bbf);

  // ---- _conv (gate -> pack cat -> so2 GEMMs -> rot_inv scatter), chunked ----
  auto run_conv = [&](ConvP p, int cin, int cout, const float* xf, float* aggout) {
    for (int e0 = 0; e0 < E; e0 += CHK) {
      int c = (E - e0 < CHK) ? (E - e0) : CHK;
      gemm(stream, eCur + (size_t)e0 * 128, 128, p.We1, 64, 128, h1c, 64, c, p.be1, 1.f, 0, 1);
      gemm(stream, h1c, 64, p.We2, cin, 64, gatec, cin, c, p.be2, 1.f, 0, 0);
      k_pack_cat_conv<<<g1((long)c * cin), 256, 0, stream>>>(xf, gatec, Dm, ei, e0, c, catc, cin);
      gemm(stream, catc,           4*cin, p.W0, 2*cout, 2*cin, y0c, 2*cout, c, nullptr,  1.f, 0, 0);
      gemm(stream, catc + 3*cin,   4*cin, p.Wr, cout,   cin,   yrc, cout,   c, nullptr,  1.f, 0, 0);
      gemm(stream, catc + 2*cin,   4*cin, p.Wi, cout,   cin,   yrc, cout,   c, nullptr, -1.f, 1, 0);
      gemm(stream, catc + 3*cin,   4*cin, p.Wi, cout,   cin,   yic, cout,   c, nullptr,  1.f, 0, 0);
      gemm(stream, catc + 2*cin,   4*cin, p.Wr, cout,   cin,   yic, cout,   c, nullptr,  1.f, 1, 0);
      k_scatter_conv<<<g1((long)c * 4 * cout), 256, 0, stream>>>(y0c, yrc, yic, Dm, ei, e0, c, aggout, cout);
    }
  };

  // ---- embed ----
  {
    ConvP pc = conv_at(6);
    k_zero<<<g1((long)N * 128), 256, 0, stream>>>(agg, (long)N * 128);
    run_conv(pc, 6, 32, bbf, agg);
    k_combine_embed<<<g1((long)N * 256), 256, 0, stream>>>(agg, latent, res);
  }

  // ---- transformer block ----
  auto run_block = [&](int b, int cout, bool residual) {
    const float *g0 = P(b), *b0 = P(b+1), *g1p = P(b+2);
    ConvP cp = conv_at(b + 3);
    const float *Wa = P(b+10), *avec = P(b+11), *Wo = P(b+12);
    const float *g0f = P(b+13), *b0f = P(b+14), *g1f = P(b+15);
    const float *W1 = P(b+16), *b1 = P(b+17), *Wg = P(b+18), *bg = P(b+19);
    const float *W1v = P(b+20), *W2 = P(b+21), *b2 = P(b+22), *W2v = P(b+23);

    k_eqnorm<<<g1(N), 256, 0, stream>>>(res, xn, g0, b0, g1p, 64);

    for (int e0 = 0; e0 < E; e0 += CHK) {
      int c = (E - e0 < CHK) ? (E - e0) : CHK;
      gemm(stream, eCur + (size_t)e0 * 128, 128, cp.We1, 64, 128, h1c, 64, c, cp.be1, 1.f, 0, 1);
      gemm(stream, h1c, 64, cp.We2, 128, 64, gatec, 128, c, cp.be2, 1.f, 0, 0);
      k_pack_cat_block<<<g1((long)c * 128), 256, 0, stream>>>(xn, gatec, Dm, ei, e0, c, catc);
      // af = silu(c0 @ Wa) -> eA chunk, then logits, then y0 overwrites eA chunk
      gemm(stream, catc, 512, Wa, 128, 128, eA + (size_t)e0 * 128, 128, c, nullptr, 1.f, 0, 1);
      k_logit<<<g1((long)c * 8), 256, 0, stream>>>(eA + (size_t)e0 * 128, avec,
                                                   logitb + (size_t)e0 * 8, c);
      gemm(stream, catc,       512, cp.W0, 128, 256, eA + (size_t)e0 * 128, 128, c, nullptr,  1.f, 0, 0);
      gemm(stream, catc + 384, 512, cp.Wr, 64,  128, eB + (size_t)e0 * 64,  64,  c, nullptr,  1.f, 0, 0);
      gemm(stream, catc + 256, 512, cp.Wi, 64,  128, eB + (size_t)e0 * 64,  64,  c, nullptr, -1.f, 1, 0);
      gemm(stream, catc + 384, 512, cp.Wi, 64,  128, eC + (size_t)e0 * 64,  64,  c, nullptr,  1.f, 0, 0);
      gemm(stream, catc + 256, 512, cp.Wr, 64,  128, eC + (size_t)e0 * 64,  64,  c, nullptr,  1.f, 1, 0);
    }

    // attention softmax over incoming edges
    k_init_stats<<<g1((long)N * 8), 256, 0, stream>>>(mxb, denb);
    k_amax<<<g1((long)E * 8), 256, 0, stream>>>(logitb, ei, mxb);
    k_expsum<<<g1((long)E * 8), 256, 0, stream>>>(logitb, ei, mxb, denb, attnb);
    k_attndiv<<<g1((long)E * 8), 256, 0, stream>>>(attnb, ei, denb);

    k_zero<<<g1((long)N * 256), 256, 0, stream>>>(agg, (long)N * 256);
    k_scatter_block<<<g1((long)E * 256), 256, 0, stream>>>(eA, eB, eC, attnb, Dm, ei, agg);

    // out = agg @ Wo  -> nA (N,4,cout)
    gemm(stream, agg, 64, Wo, cout, 64, nA, cout, 4 * N, nullptr, 1.f, 0, 0);
    if (residual)
      k_add<<<g1((long)N * 4 * cout), 256, 0, stream>>>(nA, res, (long)N * 4 * cout);

    k_eqnorm<<<g1(N), 256, 0, stream>>>(nA, xn, g0f, b0f, g1f, cout);
    // FFN
    gemm(stream, xn, 4 * cout, W1, 32, cout, nh1, 32, N, b1, 1.f, 0, 1);
    gemm(stream, xn, 4 * cout, Wg, 32, cout, ng,  32, N, bg, 1.f, 0, 1);
    gemm(stream, nh1, 32, W2, cout, 32, nhs, cout, N, b2, 1.f, 0, 0);
    gemm(stream, xn, cout, W1v, 32, cout, nB, 32, 4 * N, nullptr, 1.f, 0, 0);
    k_vvmul<<<g1((long)N * 128), 256, 0, stream>>>(nB, ng);
    gemm(stream, nB, 32, W2v, cout, 32, agg, cout, 4 * N, nullptr, 1.f, 0, 0);
    k_assemble<<<g1((long)N * 4 * cout), 256, 0, stream>>>(nA, nhs, agg, res, cout);
  };

  // ---- 4 blocks + edge updates ----
  for (int L = 0; L < 4; ++L) {
    int tb = 13 + L * 28;
    run_block(tb, 64, true);
    const float *Wu1 = P(tb+24), *bu1 = P(tb+25), *Wu2 = P(tb+26), *bu2 = P(tb+27);
    for (int e0 = 0; e0 < E; e0 += CHK) {
      int c = (E - e0 < CHK) ? (E - e0) : CHK;
      k_pack_z<<<g1((long)c * 256), 256, 0, stream>>>(eCur, res, ei, e0, c, catc);
      gemm(stream, catc, 256, Wu1, 128, 256, eA + (size_t)e0 * 128, 128, c, bu1, 1.f, 0, 1);
    }
    gemm(stream, eA, 128, Wu2, 128, 128, eCur, 128, E, bu2, 1.f, 1, 0);
  }

  // ---- project block (no residual, cout=32) ----
  run_block(125, 32, false);

  // ---- out_atoms conv ----
  {
    ConvP pc = conv_at(149);
    k_zero<<<g1((long)N * 364), 256, 0, stream>>>(atom, (long)N * 364);
    run_conv(pc, 32, 91, res, atom);
  }

  // ---- seq head + decoded ----
  {
    const float *lg = P(156), *lb = P(157), *W1 = P(158), *b1 = P(159), *W2 = P(160), *b2 = P(161);
    k_seq_ln<<<g1(N), 256, 0, stream>>>(atom, lg, lb, hh);
    gemm(stream, hh, 91, W1, 91, 91, hr, 91, N, b1, 1.f, 0, 2);
    gemm(stream, hr, 91, W2, 20, 91, sl, 20, N, b2, 1.f, 0, 0);
    float* outp = (float*)d_out;
    k_decoded<<<g1((long)N * 273), 256, 0, stream>>>(atom, outp);
    k_logsoftmax<<<g1(N), 256, 0, stream>>>(sl, outp + (size_t)N * 273);
  }
}